// ConnectLoss_34093450396259
// MI455X (gfx1250) — compile-verified
//
#include <hip/hip_runtime.h>

#define N_   4
#define C_   8
#define H_   512
#define W_   512
#define HW_  (H_*W_)
#define G_   (N_*C_)          // 32 mask images per set
#define ML_  96
#define MP_  96
#define ML1_ 97
#define BINS_ (ML1_*ML1_)     // 9409

typedef __attribute__((ext_vector_type(2))) float v2f;
typedef __attribute__((ext_vector_type(8))) float v8f;

#if __has_builtin(__builtin_amdgcn_wmma_f32_16x16x4_f32)
#define HAVE_WMMA_F32 1
#endif

// ---------------- workspace layout (bytes) ----------------
#define OFF_LABP 0ULL
#define OFF_LABL (OFF_LABP + (size_t)G_*HW_*4)
#define OFF_RK   (OFF_LABL + (size_t)G_*HW_*4)
#define OFF_PRED (OFF_RK   + (size_t)G_*HW_*4)
#define OFF_BSUM (OFF_PRED + (size_t)N_*HW_*4)
#define OFF_KP   (OFF_BSUM + (size_t)G_*1024*4)
#define OFF_KL   (OFF_KP   + (size_t)G_*4)
#define OFF_SL   (OFF_KL   + (size_t)G_*4)
#define OFF_SP   (OFF_SL   + (size_t)G_*ML1_*4)
#define OFF_SC   (OFF_SP   + (size_t)G_*ML1_*4)
#define OFF_CNT  (OFF_SC   + (size_t)G_*4)          // accumulators start here
#define OFF_MISS (OFF_CNT  + (size_t)G_*BINS_*4)
#define OFF_LCNT (OFF_MISS + (size_t)G_*4)
#define ACC_WORDS (G_*BINS_ + G_ + G_)

// ---------------- kernels ----------------
__global__ void k_zero(unsigned* p, int n) {
  int i = blockIdx.x*256 + threadIdx.x;
  if (i < n) p[i] = 0u;
}

__global__ void k_argmax(const float* __restrict__ logits, int* __restrict__ pred) {
  int p = blockIdx.x*256 + threadIdx.x;          // over N*HW
  int n = p / HW_, off = p % HW_;
  const float* base = logits + (size_t)n*C_*HW_ + off;
  float best = base[0]; int bi = 0;
  for (int c = 1; c < C_; ++c) {
    float v = base[(size_t)c*HW_];
    if (v > best) { best = v; bi = c; }          // first max wins (matches argmax)
  }
  pred[p] = bi;
}

// mask(g,p): cls[n*HW+p] == c where g=n*C+c. lab = p if mask else -1.
__global__ void k_ccl_init(const int* __restrict__ cls, int* __restrict__ lab) {
  int t = blockIdx.x*256 + threadIdx.x;          // over G*HW
  int g = t / HW_, p = t % HW_;
  int n = g / C_, c = g % C_;
  lab[t] = (cls[n*HW_ + p] == c) ? p : -1;
}

__device__ __forceinline__ int ccl_find(const int* L, int x) {
  int y = L[x];
  while (y != x) { x = y; y = L[x]; }
  return x;
}

__device__ __forceinline__ void ccl_union(int* L, int a, int b) {
  a = ccl_find(L, a);
  b = ccl_find(L, b);
  while (a != b) {
    if (a < b) { int t = a; a = b; b = t; }      // a > b
    int old = atomicMin(&L[a], b);
    a = (old == a) ? b : old;                    // Playne-Hawick reduction
  }
}

__global__ void k_ccl_merge(int* __restrict__ lab) {
  int t = blockIdx.x*256 + threadIdx.x;
  int g = t / HW_, p = t % HW_;
  int* L = lab + (size_t)g*HW_;
  if (L[p] < 0) return;
  int x = p & (W_-1), y = p >> 9;
  if (x > 0 && L[p-1] >= 0) ccl_union(L, p, p-1);
  if (y > 0) {
    int q = p - W_;
    if (L[q] >= 0)            ccl_union(L, p, q);
    if (x > 0     && L[q-1] >= 0) ccl_union(L, p, q-1);
    if (x < W_-1  && L[q+1] >= 0) ccl_union(L, p, q+1);
  }
}

__global__ void k_ccl_flatten(int* __restrict__ lab) {
  int t = blockIdx.x*256 + threadIdx.x;
  int g = t / HW_, p = t % HW_;
  int* L = lab + (size_t)g*HW_;
  if (L[p] < 0) return;
  L[p] = ccl_find(L, p);                         // root = min pixel index of comp
}

// per-block root counts: gridDim = G*1024, block = 256
__global__ void k_scanA(const int* __restrict__ lab, int* __restrict__ bsum) {
  int b = blockIdx.x;
  int g = b >> 10, p = (b & 1023)*256 + threadIdx.x;
  int isr = (lab[(size_t)g*HW_ + p] == p) ? 1 : 0;
  unsigned long long bal = __ballot(isr);
  __shared__ int wsum[8];
  int lane = threadIdx.x & 31, wv = threadIdx.x >> 5;
  if (lane == 0) wsum[wv] = __popcll(bal);
  __syncthreads();
  if (threadIdx.x == 0) {
    int s = 0;
    for (int i = 0; i < 8; ++i) s += wsum[i];
    bsum[b] = s;
  }
}

// exclusive scan of 1024 block sums per image; gridDim = G, block = 256
__global__ void k_scanB(int* __restrict__ bsum, int* __restrict__ K) {
  int g = blockIdx.x, tid = threadIdx.x;
  int* bs = bsum + g*1024;
  int v0 = bs[tid*4+0], v1 = bs[tid*4+1], v2 = bs[tid*4+2], v3 = bs[tid*4+3];
  int s = v0 + v1 + v2 + v3;
  __shared__ int sh[256];
  sh[tid] = s; __syncthreads();
  for (int off = 1; off < 256; off <<= 1) {
    int t = (tid >= off) ? sh[tid-off] : 0;
    __syncthreads();
    sh[tid] += t;
    __syncthreads();
  }
  int excl = sh[tid] - s;
  bs[tid*4+0] = excl;
  bs[tid*4+1] = excl + v0;
  bs[tid*4+2] = excl + v0 + v1;
  bs[tid*4+3] = excl + v0 + v1 + v2;
  if (tid == 255) K[g] = sh[255];
}

// inclusive raster rank of roots -> rk[g*HW+p]; gridDim = G*1024
__global__ void k_scanC(const int* __restrict__ lab, const int* __restrict__ boff,
                        int* __restrict__ rk) {
  int b = blockIdx.x;
  int g = b >> 10, p = (b & 1023)*256 + threadIdx.x;
  size_t gp = (size_t)g*HW_ + p;
  int isr = (lab[gp] == p) ? 1 : 0;
  unsigned long long bal = __ballot(isr);
  unsigned m32 = (unsigned)bal;
  int lane = threadIdx.x & 31, wv = threadIdx.x >> 5;
  int incl = __popc(m32 & (0xFFFFFFFFu >> (31 - lane)));
  __shared__ int wsum[8];
  if (lane == 31) wsum[wv] = __popc(m32);
  __syncthreads();
  int woff = 0;
  for (int i = 0; i < wv; ++i) woff += wsum[i];
  rk[gp] = boff[b] + woff + incl;
}

// overwrite lab with comp id (1..K raster order, 0 = background)
__global__ void k_comp(int* __restrict__ lab, const int* __restrict__ rk) {
  int t = blockIdx.x*256 + threadIdx.x;
  int g = t / HW_;
  int v = lab[t];
  lab[t] = (v >= 0) ? rk[(size_t)g*HW_ + v] : 0;
}

__global__ void k_hist(const int* __restrict__ compL, const int* __restrict__ compP,
                       unsigned* __restrict__ cnt) {
  int t = blockIdx.x*256 + threadIdx.x;
  int g = t / HW_;
  int cl = compL[t]; if (cl > ML_) cl = 0;
  int cp = compP[t]; if (cp > MP_) cp = 0;
  __shared__ unsigned z;
  if (threadIdx.x == 0) z = 0u;
  __syncthreads();
  if ((cl | cp) == 0) atomicAdd(&z, 1u);                 // hot background bin -> LDS
  else atomicAdd(&cnt[g*BINS_ + cl*ML1_ + cp], 1u);
  __syncthreads();
  if (threadIdx.x == 0 && z) atomicAdd(&cnt[g*BINS_], z);
}

__global__ void k_missed(const int* __restrict__ labels, const int* __restrict__ pred,
                         unsigned* __restrict__ missedCnt, unsigned* __restrict__ labCnt) {
  __shared__ unsigned ml[C_], ll[C_];
  int tid = threadIdx.x;
  if (tid < C_) { ml[tid] = 0u; ll[tid] = 0u; }
  __syncthreads();
  int p = blockIdx.x*256 + tid;                  // over N*HW, block within one n
  int n = p / HW_;
  int lc = labels[p], pc = pred[p];
  atomicAdd(&ll[lc], 1u);
  if (lc != pc) atomicAdd(&ml[lc], 1u);
  __syncthreads();
  if (tid < C_) {
    if (ml[tid]) atomicAdd(&missedCnt[n*C_ + tid], ml[tid]);
    if (ll[tid]) atomicAdd(&labCnt[n*C_ + tid],  ll[tid]);
  }
}

// size_l[i] = sum_j cnt[i][j], size_p[j] = sum_i cnt[i][j]; exact in f32 (counts < 2^24).
// One wave per g; 16x16x4 f32 WMMA with a ones operand does the reduction.
__global__ void k_sums(const unsigned* __restrict__ cnt,
                       float* __restrict__ size_l, float* __restrict__ size_p) {
  int g = blockIdx.x;
  const unsigned* M = cnt + g*BINS_;
#ifdef HAVE_WMMA_F32
  int lane = threadIdx.x;                        // 0..31, all active
  int half = lane >> 4;                          // K-pair select per A/B layout
  int l16  = lane & 15;
  // column sums -> size_p : D = ones(16x4) x B(4x16), accumulate over K
  for (int tile = 0; tile < 7; ++tile) {
    int col = tile*16 + l16;
    v8f acc = {0.f,0.f,0.f,0.f,0.f,0.f,0.f,0.f};
    for (int k0 = 0; k0 < ML1_; k0 += 4) {
      int kA = k0 + 2*half, kB = kA + 1;
      float b0 = (col < ML1_ && kA < ML1_) ? (float)M[kA*ML1_ + col] : 0.0f;
      float b1 = (col < ML1_ && kB < ML1_) ? (float)M[kB*ML1_ + col] : 0.0f;
      v2f A = {1.0f, 1.0f};
      v2f B = {b0, b1};
      acc = __builtin_amdgcn_wmma_f32_16x16x4_f32(false, A, false, B, (short)0, acc,
                                                  false, false);
    }
    if (half == 0 && col < ML1_) size_p[g*ML1_ + col] = acc[0];   // row M=0
  }
  // row sums -> size_l : D = A(16x4) x ones(4x16)
  for (int tile = 0; tile < 7; ++tile) {
    int row = tile*16 + l16;
    v8f acc = {0.f,0.f,0.f,0.f,0.f,0.f,0.f,0.f};
    for (int k0 = 0; k0 < ML1_; k0 += 4) {
      int kA = k0 + 2*half, kB = kA + 1;
      float a0 = (row < ML1_ && kA < ML1_) ? (float)M[row*ML1_ + kA] : 0.0f;
      float a1 = (row < ML1_ && kB < ML1_) ? (float)M[row*ML1_ + kB] : 0.0f;
      v2f A = {a0, a1};
      v2f B = {1.0f, 1.0f};
      acc = __builtin_amdgcn_wmma_f32_16x16x4_f32(false, A, false, B, (short)0, acc,
                                                  false, false);
    }
    if (l16 == 0) {                              // lane0 -> rows m0..m0+7, lane16 -> +8
      int base = tile*16 + 8*half;
      #pragma unroll
      for (int r = 0; r < 8; ++r) {
        int rr = base + r;
        if (rr < ML1_) size_l[g*ML1_ + rr] = acc[r];
      }
    }
  }
#else
  int lane = threadIdx.x;
  for (int j = lane; j < ML1_; j += 32) {
    float s = 0.f;
    for (int i = 0; i < ML1_; ++i) s += (float)M[i*ML1_ + j];
    size_p[g*ML1_ + j] = s;
  }
  for (int i = lane; i < ML1_; i += 32) {
    float s = 0.f;
    for (int j = 0; j < ML1_; ++j) s += (float)M[i*ML1_ + j];
    size_l[g*ML1_ + i] = s;
  }
#endif
}

__global__ void k_loss(const unsigned* __restrict__ cnt,
                       const float* __restrict__ size_l, const float* __restrict__ size_p,
                       const int* __restrict__ Kp, const int* __restrict__ Kl,
                       const unsigned* __restrict__ missedCnt,
                       const unsigned* __restrict__ labCnt,
                       float* __restrict__ sc) {
  int g = blockIdx.x, tid = threadIdx.x;         // block = 128
  int kp = Kp[g], kl = Kl[g];
  bool capped = (kp + 1) > 2*(kl + 1);
  int mn = (kp + 1) < 10 ? (kp + 1) : 10;
  int real_pred = capped ? (mn - 1) : kp;
  int real_label = kl;
  float cval = (float)(g % C_);

  __shared__ float sl[ML1_], sp[ML1_], contrib[ML_];
  __shared__ int colAny[ML1_];
  for (int j = tid; j < ML1_; j += 128) {
    sl[j] = size_l[g*ML1_ + j];
    sp[j] = size_p[g*ML1_ + j];
    colAny[j] = 0;
  }
  __syncthreads();

  if (tid < ML_) {
    int i = tid + 1;
    bool vi = tid < real_label;
    const unsigned* row = cnt + g*BINS_ + i*ML1_;
    __builtin_prefetch(row, 0, 0);
    float ps = 0.f; int pn = 0;
    for (int j = 1; j <= MP_; ++j) {
      float inter = cval * (float)row[j];
      if (vi && ((j - 1) < real_pred) && inter > 0.f) {
        float uni = cval*sp[j] + sl[i] - inter;
        ps += inter / uni;
        pn += 1;
        colAny[j] = 1;                           // idempotent
      }
    }
    contrib[tid] = (pn > 0) ? (ps / (float)pn) : 0.f;
  }
  __syncthreads();

  if (tid == 0) {                                // sequential -> deterministic
    float pcs = 0.f;
    for (int r = 0; r < ML_; ++r) pcs += contrib[r];
    int lone = 0;
    for (int j = 1; j <= MP_; ++j)
      if (((j - 1) < real_pred) && !colAny[j]) lone++;
    int den = real_label + lone; if (den < 1) den = 1;
    float img_conn = pcs / (float)den;
    float missed = (float)missedCnt[g] / (float)HW_;
    int present = labCnt[g] > 0;
    float s = (real_pred > 0) ? (1.0f - img_conn) : (missed + 1.0f);
    if (!(present && real_label > 0)) s = 0.0f;
    sc[g] = s;
  }
}

__global__ void k_final(const float* __restrict__ sc, const unsigned* __restrict__ labCnt,
                        float* __restrict__ out) {
  if (threadIdx.x == 0 && blockIdx.x == 0) {
    float tot = 0.f;
    for (int n = 0; n < N_; ++n) {
      int cn = 0; float s = 0.f;
      for (int c = 0; c < C_; ++c) {
        if (labCnt[n*C_ + c] > 0) cn++;
        s += sc[n*C_ + c];
      }
      if (cn < 1) cn = 1;
      tot += s / (float)cn;
    }
    out[0] = tot / (float)N_;
  }
}

// ---------------- launch ----------------
extern "C" void kernel_launch(void* const* d_in, const int* in_sizes, int n_in,
                              void* d_out, int out_size, void* d_ws, size_t ws_size,
                              hipStream_t stream) {
  const float* logits = (const float*)d_in[0];
  const int*   labels = (const int*)d_in[1];
  char* ws = (char*)d_ws;
  int* labP      = (int*)(ws + OFF_LABP);
  int* labL      = (int*)(ws + OFF_LABL);
  int* rk        = (int*)(ws + OFF_RK);
  int* pred      = (int*)(ws + OFF_PRED);
  int* bsum      = (int*)(ws + OFF_BSUM);
  int* Kp        = (int*)(ws + OFF_KP);
  int* Kl        = (int*)(ws + OFF_KL);
  float* size_l  = (float*)(ws + OFF_SL);
  float* size_p  = (float*)(ws + OFF_SP);
  float* sc      = (float*)(ws + OFF_SC);
  unsigned* cnt  = (unsigned*)(ws + OFF_CNT);
  unsigned* miss = (unsigned*)(ws + OFF_MISS);
  unsigned* lcnt = (unsigned*)(ws + OFF_LCNT);

  const int GHW_BLKS = (G_*HW_)/256;             // 32768
  const int NHW_BLKS = (N_*HW_)/256;             // 4096

  k_zero<<<(ACC_WORDS + 255)/256, 256, 0, stream>>>(cnt, ACC_WORDS);
  k_argmax<<<NHW_BLKS, 256, 0, stream>>>(logits, pred);

  k_ccl_init<<<GHW_BLKS, 256, 0, stream>>>(pred, labP);
  k_ccl_init<<<GHW_BLKS, 256, 0, stream>>>(labels, labL);
  k_ccl_merge<<<GHW_BLKS, 256, 0, stream>>>(labP);
  k_ccl_merge<<<GHW_BLKS, 256, 0, stream>>>(labL);
  k_ccl_flatten<<<GHW_BLKS, 256, 0, stream>>>(labP);
  k_ccl_flatten<<<GHW_BLKS, 256, 0, stream>>>(labL);

  // raster-order component numbering: pred set
  k_scanA<<<G_*1024, 256, 0, stream>>>(labP, bsum);
  k_scanB<<<G_, 256, 0, stream>>>(bsum, Kp);
  k_scanC<<<G_*1024, 256, 0, stream>>>(labP, bsum, rk);
  k_comp<<<GHW_BLKS, 256, 0, stream>>>(labP, rk);
  // label set (reuses rk/bsum)
  k_scanA<<<G_*1024, 256, 0, stream>>>(labL, bsum);
  k_scanB<<<G_, 256, 0, stream>>>(bsum, Kl);
  k_scanC<<<G_*1024, 256, 0, stream>>>(labL, bsum, rk);
  k_comp<<<GHW_BLKS, 256, 0, stream>>>(labL, rk);

  k_hist<<<GHW_BLKS, 256, 0, stream>>>(labL, labP, cnt);
  k_missed<<<NHW_BLKS, 256, 0, stream>>>(labels, pred, miss, lcnt);
  k_sums<<<G_, 32, 0, stream>>>(cnt, size_l, size_p);
  k_loss<<<G_, 128, 0, stream>>>(cnt, size_l, size_p, Kp, Kl, miss, lcnt, sc);
  k_final<<<1, 32, 0, stream>>>(sc, lcnt, (float*)d_out);
}